// FlatTreeHRRTorchComp_6528350290487
// MI455X (gfx1250) — compile-verified
//
#include <hip/hip_runtime.h>
#include <math.h>

#define HH 1024
#define DD 128
#define EE 8
#define AA 4

typedef __attribute__((ext_vector_type(2))) float v2f;
typedef __attribute__((ext_vector_type(8))) float v8f;

struct C4 { float4 re, im; };

__device__ __forceinline__ float4 f4mul(float4 a, float4 b) {
    return make_float4(a.x*b.x, a.y*b.y, a.z*b.z, a.w*b.w);
}
__device__ __forceinline__ float4 f4add(float4 a, float4 b) {
    return make_float4(a.x+b.x, a.y+b.y, a.z+b.z, a.w+b.w);
}
__device__ __forceinline__ float4 f4sub(float4 a, float4 b) {
    return make_float4(a.x-b.x, a.y-b.y, a.z-b.z, a.w-b.w);
}
__device__ __forceinline__ float4 f4scale(float s, float4 a) {
    return make_float4(s*a.x, s*a.y, s*a.z, s*a.w);
}
__device__ __forceinline__ float f4sumsq(float4 a) {
    return a.x*a.x + a.y*a.y + a.z*a.z + a.w*a.w;
}

__device__ __forceinline__ C4 czero() {
    C4 r; r.re = make_float4(0.f,0.f,0.f,0.f); r.im = make_float4(0.f,0.f,0.f,0.f); return r;
}
__device__ __forceinline__ C4 cadd(const C4& a, const C4& b) {
    C4 r; r.re = f4add(a.re, b.re); r.im = f4add(a.im, b.im); return r;
}
__device__ __forceinline__ C4 cscale(float s, const C4& a) {
    C4 r; r.re = f4scale(s, a.re); r.im = f4scale(s, a.im); return r;
}
// complex elementwise multiply: (xr*yr - xi*yi, xr*yi + xi*yr)
__device__ __forceinline__ C4 cmul(const C4& x, const C4& y) {
    C4 r;
    r.re = f4sub(f4mul(x.re, y.re), f4mul(x.im, y.im));
    r.im = f4add(f4mul(x.re, y.im), f4mul(x.im, y.re));
    return r;
}
__device__ __forceinline__ float csumsq(const C4& x) {
    return f4sumsq(x.re) + f4sumsq(x.im);
}
__device__ __forceinline__ C4 loadC(const float* p, int h4) {
    C4 r;
    r.re = ((const float4*)p)[h4];
    r.im = ((const float4*)(p + HH))[h4];
    return r;
}
__device__ __forceinline__ void storeC(float* p, int h4, const C4& v) {
    ((float4*)p)[h4] = v.re;
    ((float4*)(p + HH))[h4] = v.im;
}

// 32-lane sum via V_WMMA_F32_16X16X4_F32.
// A: lane l puts partial in VGPR0 (K=0 for lanes 0-15 -> A[m,0]=p_m,
//    K=2 for lanes 16-31 -> A[m,2]=p_{m+16}); VGPR1 = 0.
// B: all ones. D[m,n] = p_m + p_{m+16}.
// Sum of the 8 D VGPRs gives lanes 0-15: sum(p_0..7,p_16..23),
// lanes 16-31: sum(p_8..15,p_24..31); shfl_xor(16) completes the sum.
__device__ __forceinline__ float waveSum(float v) {
    v2f a; a[0] = v;   a[1] = 0.0f;
    v2f b; b[0] = 1.0f; b[1] = 1.0f;
    v8f c = {};
    v8f d = __builtin_amdgcn_wmma_f32_16x16x4_f32(
        false, a, false, b, (short)0, c, false, false);
    float h = ((d[0] + d[1]) + (d[2] + d[3])) + ((d[4] + d[5]) + (d[6] + d[7]));
    h += __shfl_xor(h, 16, 32);
    return h;
}

// Block-wide (256 threads = 8 waves) sum, broadcast to all threads.
__device__ __forceinline__ float blockSum(float v, float* red) {
    float w = waveSum(v);
    __syncthreads();                       // protect previous reduction's reads
    if ((threadIdx.x & 31) == 0) red[threadIdx.x >> 5] = w;
    __syncthreads();
    float s = 0.f;
#pragma unroll
    for (int i = 0; i < 8; ++i) s += red[i];
    return s;
}

// nz(): normalize over the (2,H) slice owned by this block
__device__ __forceinline__ C4 nzC(const C4& x, float* red) {
    float n = blockSum(csumsq(x), red);
    float inv = 1.0f / sqrtf(n);
    return cscale(inv, x);
}

// ground_step: spec = nz(var*noise); return nz(r0*parent + r1*spec)
__device__ __forceinline__ C4 gstep(const C4& parent, const C4& var,
                                    const float* noise, float r0, float r1,
                                    int h4, float* red) {
    C4 s;
    s.re = f4mul(var.re, ((const float4*)noise)[h4]);
    s.im = f4mul(var.im, ((const float4*)(noise + HH))[h4]);
    float n1 = blockSum(csumsq(s), red);
    float i1 = r1 / sqrtf(n1);
    C4 t;
    t.re = f4add(f4scale(r0, parent.re), f4scale(i1, s.re));
    t.im = f4add(f4scale(r0, parent.im), f4scale(i1, s.im));
    return nzC(t, red);
}

// ---------------- kernel 0: per-dispatch constants -> pre[] ----------------
// pre layout (floats): [0]=nz(fixed_var), [2H]=fb, [4H]=ab, [6H]=gvL, [8H]=gvR, [10H]=roleobj
__global__ __launch_bounds__(256) void k0_pre(
    const float* fixed_var, const float* fixed_func, const float* fixed_arg,
    const float* fixed_left, const float* fixed_right, const float* fixed_disjrole,
    const float* var_Func0, const float* var_Arg0, const float* var_Left0,
    const float* var_Right0, const float* var_DisjRole0,
    const float* rF0, const float* rA0, const float* rL0,
    const float* rR0, const float* rD0,
    const float* noise_func_base, const float* noise_arg_base,
    const float* noise_left, const float* noise_right, const float* noise_role,
    float* pre)
{
    __shared__ float red[8];
    const int h4 = threadIdx.x;

    C4 t = nzC(loadC(fixed_var, h4), red);
    storeC(pre + 0, h4, t);

    t = nzC(loadC(fixed_func, h4), red);
    t = gstep(t, loadC(var_Func0, h4), noise_func_base, rF0[0], rF0[1], h4, red);
    storeC(pre + 2*HH, h4, t);

    t = nzC(loadC(fixed_arg, h4), red);
    t = gstep(t, loadC(var_Arg0, h4), noise_arg_base, rA0[0], rA0[1], h4, red);
    storeC(pre + 4*HH, h4, t);

    t = nzC(loadC(fixed_left, h4), red);
    t = gstep(t, loadC(var_Left0, h4), noise_left, rL0[0], rL0[1], h4, red);
    storeC(pre + 6*HH, h4, t);

    t = nzC(loadC(fixed_right, h4), red);
    t = gstep(t, loadC(var_Right0, h4), noise_right, rR0[0], rR0[1], h4, red);
    storeC(pre + 8*HH, h4, t);

    t = nzC(loadC(fixed_disjrole, h4), red);
    t = gstep(t, loadC(var_DisjRole0, h4), noise_role, rD0[0], rD0[1], h4, red);
    storeC(pre + 10*HH, h4, t);
}

// ---------------- kernel 1: one block per (d,e); fused eqv ----------------
__global__ __launch_bounds__(256) void k1_eqv(
    const float* pre,
    const float* noise_vars, const float* noise_argidx,
    const float* noise_funcname, const float* noise_argname,
    const float* var_Var0, const float* var_Func1,
    const float* var_Arg1, const float* var_Arg2,
    const float* rV0, const float* rF1, const float* rA1, const float* rA2,
    const float* w_func, const float* w_eq,
    float* eqv_out)
{
    __shared__ float red[8];
    const int h4 = threadIdx.x;
    const int de = blockIdx.x;   // d*EE + e

    const float rv0 = rV0[0],  rv1 = rV0[1];
    const float rf0 = rF1[0],  rf1 = rF1[1];
    const float ra10 = rA1[0], ra11 = rA1[1];
    const float ra20 = rA2[0], ra21 = rA2[1];
    const float wf0 = w_func[0], wf1 = w_func[1], wf2 = w_func[2];

    const C4 Pvar   = loadC(pre, h4);
    const C4 fb     = loadC(pre + 2*HH, h4);
    const C4 ab     = loadC(pre + 4*HH, h4);
    const C4 vVar0  = loadC(var_Var0, h4);
    const C4 vFunc1 = loadC(var_Func1, h4);
    const C4 vArg1  = loadC(var_Arg1, h4);
    const C4 vArg2  = loadC(var_Arg2, h4);

    C4 tj[2];
#pragma unroll
    for (int j = 0; j < 2; ++j) {
        const float* nfn = noise_funcname + (size_t)(de*2 + j) * 2 * HH;
        const float* nag = noise_argname  + (size_t)(de*2 + j) * 2 * HH;
        C4 fname = gstep(fb, vFunc1, nfn, rf0, rf1, h4, red);
        C4 aname = gstep(ab, vArg1, nag, ra10, ra11, h4, red);

        C4 accA = czero(), accV = czero();
        for (int a = 0; a < AA; ++a) {
            const float* nv = noise_vars   + (size_t)((de*2 + j)*AA + a) * 2 * HH;
            const float* ni = noise_argidx + (size_t)((de*2 + j)*AA + a) * 2 * HH;
            C4 vleaf = gstep(Pvar,  vVar0, nv, rv0,  rv1,  h4, red);
            C4 aidx  = gstep(aname, vArg2, ni, ra20, ra21, h4, red);
            accA = cadd(accA, cmul(aidx, vleaf));
            accV = cadd(accV, vleaf);
        }
        C4 ma = nzC(accA, red);
        C4 mv = nzC(accV, red);
        C4 f  = cadd(cadd(cscale(wf0, fname), cscale(wf1, ma)), cscale(wf2, mv));
        tj[j] = nzC(f, red);
    }

    const C4 gvL = loadC(pre + 6*HH, h4);
    const C4 gvR = loadC(pre + 8*HH, h4);
    const float we0 = w_eq[0], we1 = w_eq[1], we2 = w_eq[2];
    const float we3 = w_eq[3], we4 = w_eq[4], we5 = w_eq[5];

    C4 lo = cmul(gvL, tj[0]);
    C4 ro = cmul(gvR, tj[1]);
    C4 acc = cscale(we0, lo);
    acc = cadd(acc, cscale(we1, ro));
    acc = cadd(acc, cscale(we2, cmul(lo, ro)));
    acc = cadd(acc, cscale(we3, tj[0]));
    acc = cadd(acc, cscale(we4, tj[1]));
    acc = cadd(acc, cscale(we5, cmul(tj[1], tj[1])));
    C4 eq = nzC(acc, red);
    storeC(eqv_out + (size_t)de * 2 * HH, h4, eq);
}

// ---------------- kernel 2: per d: sum over E, mo/mvv/disjv ----------------
__global__ __launch_bounds__(256) void k2_disj(
    const float* pre, const float* eqv, const float* w_disj, float* disj_out)
{
    __shared__ float red[8];
    const int h4 = threadIdx.x;
    const int d = blockIdx.x;

    C4 S = czero();
    for (int e = 0; e < EE; ++e)
        S = cadd(S, loadC(eqv + (size_t)(d*EE + e) * 2 * HH, h4));

    const C4 role = loadC(pre + 10*HH, h4);
    // sum_e assoc(role, eqv) == assoc(role, sum_e eqv) (role constant over e)
    C4 mo  = nzC(cmul(role, S), red);
    C4 mvv = nzC(S, red);

    const float wd0 = w_disj[0], wd1 = w_disj[1], wd2 = w_disj[2];
    C4 t = cadd(cadd(cscale(wd0, cmul(mo, mo)), cscale(wd1, mo)), cscale(wd2, mvv));
    C4 dis = nzC(t, red);
    storeC(disj_out + (size_t)d * 2 * HH, h4, dis);
}

// ---------------- kernel 3: sum over D, final output ----------------
__global__ __launch_bounds__(256) void k3_out(
    const float* disj, const float* w_conj, float* out)
{
    __shared__ float red[8];
    const int h4 = threadIdx.x;

    C4 S = czero();
    for (int d = 0; d < DD; ++d)
        S = cadd(S, loadC(disj + (size_t)d * 2 * HH, h4));

    C4 mc = nzC(S, red);
    const float wc0 = w_conj[0], wc1 = w_conj[1];
    C4 t = cadd(cscale(wc0, cmul(mc, mc)), cscale(wc1, mc));
    C4 o = nzC(t, red);
    storeC(out, h4, o);
}

extern "C" void kernel_launch(void* const* d_in, const int* in_sizes, int n_in,
                              void* d_out, int out_size, void* d_ws, size_t ws_size,
                              hipStream_t stream) {
    // setup_inputs() order
    const float* fixed_var       = (const float*)d_in[0];
    const float* fixed_func      = (const float*)d_in[1];
    const float* fixed_arg       = (const float*)d_in[2];
    const float* fixed_left      = (const float*)d_in[3];
    const float* fixed_right     = (const float*)d_in[4];
    const float* fixed_disjrole  = (const float*)d_in[5];
    const float* var_Var0        = (const float*)d_in[6];
    const float* var_Func0       = (const float*)d_in[7];
    const float* var_Func1       = (const float*)d_in[8];
    const float* var_Arg0        = (const float*)d_in[9];
    const float* var_Arg1        = (const float*)d_in[10];
    const float* var_Arg2        = (const float*)d_in[11];
    const float* var_Left0       = (const float*)d_in[12];
    const float* var_Right0      = (const float*)d_in[13];
    const float* var_DisjRole0   = (const float*)d_in[14];
    const float* ratio_Var0      = (const float*)d_in[15];
    const float* ratio_Func0     = (const float*)d_in[16];
    const float* ratio_Func1     = (const float*)d_in[17];
    const float* ratio_Arg0      = (const float*)d_in[18];
    const float* ratio_Arg1      = (const float*)d_in[19];
    const float* ratio_Arg2      = (const float*)d_in[20];
    const float* ratio_Left0     = (const float*)d_in[21];
    const float* ratio_Right0    = (const float*)d_in[22];
    const float* ratio_DisjRole0 = (const float*)d_in[23];
    const float* w_func          = (const float*)d_in[24];
    const float* w_eq            = (const float*)d_in[25];
    const float* w_disj          = (const float*)d_in[26];
    const float* w_conj          = (const float*)d_in[27];
    const float* noise_vars      = (const float*)d_in[28];
    const float* noise_argidx    = (const float*)d_in[29];
    const float* noise_funcname  = (const float*)d_in[30];
    const float* noise_argname   = (const float*)d_in[31];
    const float* noise_func_base = (const float*)d_in[32];
    const float* noise_arg_base  = (const float*)d_in[33];
    const float* noise_left      = (const float*)d_in[34];
    const float* noise_right     = (const float*)d_in[35];
    const float* noise_role      = (const float*)d_in[36];

    float* ws   = (float*)d_ws;
    float* pre  = ws;                                  // 12*HH floats
    float* eqv  = ws + 12 * HH;                        // DD*EE*2*HH floats (8 MiB)
    float* disj = eqv + (size_t)DD * EE * 2 * HH;      // DD*2*HH floats (1 MiB)

    k0_pre<<<1, 256, 0, stream>>>(
        fixed_var, fixed_func, fixed_arg, fixed_left, fixed_right, fixed_disjrole,
        var_Func0, var_Arg0, var_Left0, var_Right0, var_DisjRole0,
        ratio_Func0, ratio_Arg0, ratio_Left0, ratio_Right0, ratio_DisjRole0,
        noise_func_base, noise_arg_base, noise_left, noise_right, noise_role,
        pre);

    k1_eqv<<<DD * EE, 256, 0, stream>>>(
        pre, noise_vars, noise_argidx, noise_funcname, noise_argname,
        var_Var0, var_Func1, var_Arg1, var_Arg2,
        ratio_Var0, ratio_Func1, ratio_Arg1, ratio_Arg2,
        w_func, w_eq, eqv);

    k2_disj<<<DD, 256, 0, stream>>>(pre, eqv, w_disj, disj);

    k3_out<<<1, 256, 0, stream>>>(disj, w_conj, (float*)d_out);
}